// RootCauseAttention_18399639896424
// MI455X (gfx1250) — compile-verified
//
#include <hip/hip_runtime.h>
#include <math.h>

typedef __attribute__((ext_vector_type(2))) float v2f;
typedef __attribute__((ext_vector_type(8))) float v8f;

#define HDIM   128
#define TILE_M 16
#define WPB    8      // waves per block (1 tile per wave)
#define APITCH 132    // padded LDS row pitch in floats (128 + 4 -> conflict-free)

// ---------------------------------------------------------------------------
// Kernel 1: fused 3-way matvec via V_WMMA_F32_16X16X4_F32.
// B (128x16): col0 = W_src, col1 = W_dst, col2 = W_node, cols 3..15 = 0.
// Each wave computes a 16x16 D tile = h[16 rows] x B; cols 0/1/2 of D are the
// three per-node scores. combined[] is initialized to s_self + b_node here.
// ---------------------------------------------------------------------------
__global__ __launch_bounds__(256)
void rca_scores_wmma(const float* __restrict__ h,
                     const float* __restrict__ W_edge,
                     const float* __restrict__ W_node,
                     const float* __restrict__ b_node,
                     float* __restrict__ s_src,
                     float* __restrict__ s_dst,
                     float* __restrict__ combined,
                     int n)
{
    __shared__ float Bs[HDIM * 16];            // 8 KB, shared by all waves
    __shared__ float As[WPB][TILE_M * APITCH]; // 8 x 8.25 KB staging tiles

    const int tid  = threadIdx.x;
    const int wave = tid >> 5;
    const int lane = tid & 31;

    // Stage B: 128x16, only 3 useful columns.
    for (int i = tid; i < HDIM * 16; i += 256) {
        const int k = i >> 4, c = i & 15;
        float v = 0.0f;
        if (c == 0)      v = W_edge[k];
        else if (c == 1) v = W_edge[HDIM + k];
        else if (c == 2) v = W_node[k];
        Bs[i] = v;
    }

    const int tile   = blockIdx.x * WPB + wave;
    const int row0   = tile * TILE_M;
    const bool active = row0 < n;   // uniform per wave -> EXEC all-ones inside

    // Stage A tile (16 rows x 128 cols) with coalesced float4 loads.
    if (active) {
        const float4* hg = reinterpret_cast<const float4*>(h);
        for (int i = lane; i < TILE_M * (HDIM / 4); i += 32) {
            const int r = i >> 5;    // 32 float4 per row
            const int c = i & 31;
            float4 v = hg[(size_t)(row0 + r) * (HDIM / 4) + c];
            *reinterpret_cast<float4*>(&As[wave][r * APITCH + c * 4]) = v;
        }
    }
    __syncthreads();

    v8f acc = {};
    if (active) {
        // A 16x4 f32 layout: lanes 0-15 hold M=lane, K=k0..k0+1;
        //                    lanes 16-31 hold M=lane-16, K=k0+2..k0+3.
        // B 4x16 f32 layout mirrors with N across lanes.
        const int rA   = lane & 15;
        const int koff = (lane < 16) ? 0 : 2;
        const int nB   = lane & 15;
        for (int k0 = 0; k0 < HDIM; k0 += 4) {
            const int kA = k0 + koff;
            v2f a = *reinterpret_cast<const v2f*>(&As[wave][rA * APITCH + kA]);
            v2f b;
            b.x = Bs[kA * 16 + nB];
            b.y = Bs[(kA + 1) * 16 + nB];
            acc = __builtin_amdgcn_wmma_f32_16x16x4_f32(
                false, a, false, b, (short)0, acc, false, false);
        }
        // D layout: VGPR r holds M=r (lanes 0-15) / M=r+8 (lanes 16-31), N=lane&15
        const float bn    = b_node[0];
        const int   nCol  = lane & 15;
        const int   mBase = (lane < 16) ? 0 : 8;
        if (nCol < 3) {
            #pragma unroll
            for (int r = 0; r < 8; ++r) {
                const int row = row0 + mBase + r;
                if (row < n) {
                    const float v = acc[r];
                    if (nCol == 0)      s_src[row]    = v;
                    else if (nCol == 1) s_dst[row]    = v;
                    else                combined[row] = v + bn;
                }
            }
        }
    }
}

// ---------------------------------------------------------------------------
// Kernel 2: edge gather + fp32 atomic segment-sum into combined[dst].
// 200KB target array is L2-resident; unsafeAtomicAdd -> global_atomic_add_f32.
// ---------------------------------------------------------------------------
__global__ __launch_bounds__(256)
void rca_edges(const int* __restrict__ src,
               const int* __restrict__ dst,
               const float* __restrict__ s_src,
               const float* __restrict__ s_dst,
               const float* __restrict__ b_edge,
               float* __restrict__ combined,
               int ne)
{
    const float be = b_edge[0];
    const int stride = gridDim.x * blockDim.x;
    for (int e = blockIdx.x * blockDim.x + threadIdx.x; e < ne; e += stride) {
        const int s = src[e];
        const int d = dst[e];
        const float sc = s_src[s] + s_dst[d] + be;
        unsafeAtomicAdd(&combined[d], sc);
    }
}

// ---------------------------------------------------------------------------
// Softmax: 3-pass online (max, sum-of-exp) reduction, wave32 shuffles.
// ---------------------------------------------------------------------------
__device__ __forceinline__ void sm_merge(float& m, float& s, float m2, float s2)
{
    const float M = fmaxf(m, m2);
    s = s * __expf(m - M) + s2 * __expf(m2 - M);
    m = M;
}

__global__ __launch_bounds__(256)
void softmax_pass1(const float* __restrict__ x,
                   float* __restrict__ pm, float* __restrict__ ps, int n)
{
    __shared__ float smx[8], sms[8];
    float m = -3.0e38f, s = 0.0f;
    const int stride = gridDim.x * blockDim.x;
    for (int i = blockIdx.x * blockDim.x + threadIdx.x; i < n; i += stride) {
        const float v = x[i];
        const float M = fmaxf(m, v);
        s = s * __expf(m - M) + __expf(v - M);
        m = M;
    }
    for (int off = 16; off > 0; off >>= 1)
        sm_merge(m, s, __shfl_xor(m, off, 32), __shfl_xor(s, off, 32));
    const int lane = threadIdx.x & 31, wave = threadIdx.x >> 5;
    if (lane == 0) { smx[wave] = m; sms[wave] = s; }
    __syncthreads();
    if (threadIdx.x == 0) {
        float M = smx[0], S = sms[0];
        for (int w = 1; w < (int)(blockDim.x >> 5); ++w)
            sm_merge(M, S, smx[w], sms[w]);
        pm[blockIdx.x] = M;
        ps[blockIdx.x] = S;
    }
}

__global__ void softmax_pass2(const float* __restrict__ pm,
                              const float* __restrict__ ps,
                              float* __restrict__ MS, int nb)
{
    float m = -3.0e38f, s = 0.0f;
    for (int i = threadIdx.x; i < nb; i += 32)
        sm_merge(m, s, pm[i], ps[i]);
    for (int off = 16; off > 0; off >>= 1)
        sm_merge(m, s, __shfl_xor(m, off, 32), __shfl_xor(s, off, 32));
    if (threadIdx.x == 0) { MS[0] = m; MS[1] = s; }
}

__global__ __launch_bounds__(256)
void softmax_pass3(const float* __restrict__ x,
                   const float* __restrict__ MS,
                   float* __restrict__ out, int n)
{
    const float M    = MS[0];
    const float invS = 1.0f / MS[1];
    const int i = blockIdx.x * blockDim.x + threadIdx.x;
    if (i < n) out[i] = __expf(x[i] - M) * invS;
}

// ---------------------------------------------------------------------------
extern "C" void kernel_launch(void* const* d_in, const int* in_sizes, int n_in,
                              void* d_out, int out_size, void* d_ws, size_t ws_size,
                              hipStream_t stream)
{
    const float* h      = (const float*)d_in[0];
    const int*   ei     = (const int*)  d_in[1];   // (2, E): row0 = src, row1 = dst
    const float* W_edge = (const float*)d_in[2];   // (2H,)
    const float* b_edge = (const float*)d_in[3];   // (1,)
    const float* W_node = (const float*)d_in[4];   // (H,)
    const float* b_node = (const float*)d_in[5];   // (1,)
    float* out = (float*)d_out;

    const int n  = in_sizes[0] / HDIM;
    const int ne = in_sizes[1] / 2;
    const int* src = ei;
    const int* dst = ei + (size_t)ne;

    // Workspace layout (floats): s_src[n] | s_dst[n] | combined[n] | pm[NB] | ps[NB] | MS[2]
    const int NB = 64;
    float* ws       = (float*)d_ws;
    float* s_src    = ws;
    float* s_dst    = ws + (size_t)n;
    float* combined = ws + 2 * (size_t)n;
    float* pm       = ws + 3 * (size_t)n;
    float* ps       = pm + NB;
    float* MS       = ps + NB;

    const int tiles = (n + TILE_M - 1) / TILE_M;
    const int gA    = (tiles + WPB - 1) / WPB;
    rca_scores_wmma<<<gA, 256, 0, stream>>>(h, W_edge, W_node, b_node,
                                            s_src, s_dst, combined, n);

    rca_edges<<<2048, 256, 0, stream>>>(src, dst, s_src, s_dst, b_edge,
                                        combined, ne);

    softmax_pass1<<<NB, 256, 0, stream>>>(combined, pm, ps, n);
    softmax_pass2<<<1, 32, 0, stream>>>(pm, ps, MS, NB);
    softmax_pass3<<<(n + 255) / 256, 256, 0, stream>>>(combined, MS, out, n);
}